// SequenceLoss_44959717655232
// MI455X (gfx1250) — compile-verified
//
#include <hip/hip_runtime.h>

// ---------------------------------------------------------------------------
// Problem constants (match the reference)
// ---------------------------------------------------------------------------
#define FILLV (-100)
constexpr int Bb = 512;
constexpr int Tt = 8;
constexpr int Ss = 8192;
constexpr int Cc = 3;

constexpr int THREADS = 256;              // 8 wave32s per block
constexpr int SCHUNK  = 1024;             // s-positions per block
constexpr int SBLK    = Ss / SCHUNK;      // 8 chunks per batch row
constexpr int NBLOCKS = Bb * SBLK;        // 4096 blocks
constexpr int PER_THR = SCHUNK / THREADS; // 4 positions per thread

// ---------------------------------------------------------------------------
// CDNA5 async global->LDS path (ASYNCcnt), guarded so compile never breaks
// ---------------------------------------------------------------------------
typedef int v4i __attribute__((vector_size(16)));

#if defined(__gfx1250__)
#if __has_builtin(__builtin_amdgcn_global_load_async_to_lds_b128)
#define USE_ASYNC_LDS 1
#endif
#if __has_builtin(__builtin_amdgcn_s_wait_asynccnt)
#define WAIT_ASYNC_0() __builtin_amdgcn_s_wait_asynccnt(0)
#else
#define WAIT_ASYNC_0() asm volatile("s_wait_asynccnt 0" ::: "memory")
#endif
#endif

// ---------------------------------------------------------------------------
// Stage 1: per-block partial (sum of NLL, count of valid positions)
// ---------------------------------------------------------------------------
__global__ __launch_bounds__(THREADS) void seqloss_stage1(
    const float* __restrict__ preds,    // [B, C, S] f32
    const int*   __restrict__ classes,  // [B, T]    i32
    const int*   __restrict__ mask,     // [B, T, S] i32 (0/1)
    float* __restrict__ psum,           // [NBLOCKS]
    float* __restrict__ pcnt)           // [NBLOCKS]
{
    __shared__ int   lmask[Tt * SCHUNK];   // 32 KiB mask tile
    __shared__ float red[THREADS];

    const int blk = blockIdx.x;
    const int b   = blk >> 3;                    // blk / SBLK
    const int s0  = (blk & (SBLK - 1)) * SCHUNK; // chunk origin in S
    const int tid = threadIdx.x;

    const int* mbase = mask + ((size_t)b * Tt) * Ss + s0;

    // ---- Stream the 8 x 1024 int32 mask tile into LDS -----------------
#if USE_ASYNC_LDS
    // 2048 x 16B chunks; 8 per thread. Row t = 256 chunks of that mask row.
#pragma unroll
    for (int k = 0; k < (Tt * SCHUNK / 4) / THREADS; ++k) {
        const int idx = k * THREADS + tid;   // 0..2047
        const int t   = idx >> 8;            // row (0..7)
        const int j   = idx & 255;           // 16B chunk within row
        const int* g  = mbase + t * Ss + j * 4;
        __builtin_amdgcn_global_load_async_to_lds_b128(
            (__attribute__((address_space(1))) v4i*)(unsigned long long)g,
            (__attribute__((address_space(3))) v4i*)(&lmask[idx * 4]),
            0, 0);
    }
    WAIT_ASYNC_0();        // drain this wave's ASYNCcnt
    __syncthreads();       // other waves' tiles visible too
#else
    for (int idx = tid; idx < Tt * SCHUNK / 4; idx += THREADS) {
        const int t = idx >> 8;
        const int j = idx & 255;
        ((int4*)lmask)[idx] = *(const int4*)(mbase + t * Ss + j * 4);
    }
    __syncthreads();
#endif

    // ---- Per-batch target classes (uniform; compiler uses SMEM loads) -
    int cls[Tt];
#pragma unroll
    for (int t = 0; t < Tt; ++t) cls[t] = classes[b * Tt + t];

    const float* pbase = preds + ((size_t)b * Cc) * Ss + s0;

    // ---- Compute: stride-THREADS lane mapping => conflict-free LDS ----
    float sum = 0.0f;
    float cnt = 0.0f;
#pragma unroll
    for (int i = 0; i < PER_THR; ++i) {
        const int p = i * THREADS + tid;     // 0..1023

        // max over t of (valid ? class : FILL)  — matches reference jnp.max
        int tgt = FILLV;
#pragma unroll
        for (int t = 0; t < Tt; ++t) {
            const int m    = lmask[t * SCHUNK + p];
            const int cand = ((m == 1) && (cls[t] != FILLV)) ? cls[t] : FILLV;
            tgt = (cand > tgt) ? cand : tgt;
        }

        if (tgt != FILLV) {
            const float a0 = pbase[0 * Ss + p];
            const float a1 = pbase[1 * Ss + p];
            const float a2 = pbase[2 * Ss + p];
            const float mx  = fmaxf(a0, fmaxf(a1, a2));
            const float lse = mx + __logf(__expf(a0 - mx) +
                                          __expf(a1 - mx) +
                                          __expf(a2 - mx));
            const float xt = (tgt == 0) ? a0 : ((tgt == 1) ? a1 : a2);
            sum += lse - xt;   // -log_softmax[target]
            cnt += 1.0f;
        }
    }

    // ---- Deterministic block tree reduction ---------------------------
    red[tid] = sum;
    __syncthreads();
    for (int off = THREADS / 2; off > 0; off >>= 1) {
        if (tid < off) red[tid] += red[tid + off];
        __syncthreads();
    }
    if (tid == 0) psum[blk] = red[0];
    __syncthreads();

    red[tid] = cnt;
    __syncthreads();
    for (int off = THREADS / 2; off > 0; off >>= 1) {
        if (tid < off) red[tid] += red[tid + off];
        __syncthreads();
    }
    if (tid == 0) pcnt[blk] = red[0];
}

// ---------------------------------------------------------------------------
// Stage 2: fold 4096 partials in a fixed order -> scalar mean
// ---------------------------------------------------------------------------
__global__ __launch_bounds__(THREADS) void seqloss_stage2(
    const float* __restrict__ psum,
    const float* __restrict__ pcnt,
    float* __restrict__ out)
{
    __shared__ float rs[THREADS];
    __shared__ float rc[THREADS];
    const int tid = threadIdx.x;

    float s = 0.0f, c = 0.0f;
    for (int i = tid; i < NBLOCKS; i += THREADS) {   // fixed order per lane
        s += psum[i];
        c += pcnt[i];
    }
    rs[tid] = s;
    rc[tid] = c;
    __syncthreads();
    for (int off = THREADS / 2; off > 0; off >>= 1) {
        if (tid < off) {
            rs[tid] += rs[tid + off];
            rc[tid] += rc[tid + off];
        }
        __syncthreads();
    }
    if (tid == 0) out[0] = rs[0] / fmaxf(rc[0], 1.0f);
}

// ---------------------------------------------------------------------------
// Entry point
// ---------------------------------------------------------------------------
extern "C" void kernel_launch(void* const* d_in, const int* in_sizes, int n_in,
                              void* d_out, int out_size, void* d_ws, size_t ws_size,
                              hipStream_t stream) {
    (void)in_sizes; (void)n_in; (void)out_size; (void)ws_size;

    const float* preds   = (const float*)d_in[0];  // [B,C,S] f32
    const int*   classes = (const int*)  d_in[1];  // [B,T]   i32
    const int*   mask    = (const int*)  d_in[2];  // [B,T,S] i32

    float* psum = (float*)d_ws;            // NBLOCKS floats
    float* pcnt = psum + NBLOCKS;          // NBLOCKS floats (32 KiB total)

    seqloss_stage1<<<NBLOCKS, THREADS, 0, stream>>>(preds, classes, mask, psum, pcnt);
    seqloss_stage2<<<1, THREADS, 0, stream>>>(psum, pcnt, (float*)d_out);
}